// SGConvNet_53815940219575
// MI455X (gfx1250) — compile-verified
//
#include <hip/hip_runtime.h>

// SGConv (K=3) + linear for MI455X (gfx1250, wave32).
//  - Propagation: L2-resident scatter/gather with f32 global atomics
//    (x matrix = 25.6MB << 192MB L2, so all random traffic stays on-die).
//  - Final linear: V_WMMA_F32_16X16X4_F32, one 16x16 tile per wave.

#define EMB 64

typedef __attribute__((ext_vector_type(2))) float v2f;
typedef __attribute__((ext_vector_type(8))) float v8f;

// deg[i] = 1 (self loop)
__global__ void k_deg_init(float* __restrict__ deg, int n) {
    int i = blockIdx.x * blockDim.x + threadIdx.x;
    if (i < n) deg[i] = 1.0f;
}

// deg[col[e]] += 1 over E edges
__global__ void k_deg_count(const int* __restrict__ col, float* __restrict__ deg, int e) {
    int i = blockIdx.x * blockDim.x + threadIdx.x;
    if (i < e) atomicAdd(&deg[col[i]], 1.0f);
}

// deg -> dinv = rsqrt(deg) in place
__global__ void k_dinv(float* __restrict__ deg, int n) {
    int i = blockIdx.x * blockDim.x + threadIdx.x;
    if (i < n) {
        float d = deg[i];
        deg[i] = (d > 0.0f) ? rsqrtf(d) : 0.0f;
    }
}

// Self-loop term: xout[i] = xin[src(i)] * dinv[i]^2  (fully overwrites xout).
// xi != nullptr only on hop 1 (embedding gather; identity in the reference).
// One wave per node; each lane owns 2 consecutive features (float2).
__global__ void k_hop_init(const float* __restrict__ xin, const int* __restrict__ xi,
                           const float* __restrict__ dinv, float* __restrict__ xout, int n) {
    int t = blockIdx.x * blockDim.x + threadIdx.x;
    int node = t >> 5;
    if (node >= n) return;
    int f2 = (t & 31) << 1;
    int src = xi ? xi[node] : node;
    float w = dinv[node];
    w *= w;
    const float2 v = *(const float2*)(xin + (size_t)src * EMB + f2);
    float2* q = (float2*)(xout + (size_t)node * EMB + f2);
    *q = make_float2(v.x * w, v.y * w);
}

// Edge scatter: xout[col] += xin[src(row)] * dinv[row]*dinv[col].
// One wave per edge; each lane handles 2 features -> 2 f32 atomics/lane.
// All gathers/scatters hit L2 (feature matrix is L2-resident).
__global__ void k_hop_edges(const int* __restrict__ row, const int* __restrict__ col,
                            const float* __restrict__ dinv, const float* __restrict__ xin,
                            const int* __restrict__ xi, float* __restrict__ xout, int e) {
    int t = blockIdx.x * blockDim.x + threadIdx.x;
    int edge = t >> 5;
    if (edge >= e) return;
    int f2 = (t & 31) << 1;
    int r = row[edge];
    int c = col[edge];
    float w = dinv[r] * dinv[c];
    int src = xi ? xi[r] : r;
    const float2 v = *(const float2*)(xin + (size_t)src * EMB + f2);
    float* q = xout + (size_t)c * EMB + f2;
    atomicAdd(q,     v.x * w);
    atomicAdd(q + 1, v.y * w);
}

// out = x @ lin_w^T + lin_b  via V_WMMA_F32_16X16X4_F32.
// Block = 128 threads = 4 waves; block covers 16 nodes x 64 outputs,
// each wave computes one 16x16 tile, accumulating K=64 in 16 WMMA steps.
// A layout (16x4 f32): lane L -> M = L&15; lanes 0-15 carry K={k,k+1},
// lanes 16-31 carry K={k+2,k+3}. B = lin_w^T (B[k][n] = lin_w[n*64+k]),
// same K split across lane halves, N = L&15.
// C/D layout: VGPR j -> M = j + 8*(L>>4), N = L&15.
__global__ void k_linear_wmma(const float* __restrict__ x, const float* __restrict__ w,
                              const float* __restrict__ bias, float* __restrict__ out, int n) {
    int lane = threadIdx.x & 31;
    int wave = threadIdx.x >> 5;          // 0..3 -> output-channel tile
    int mtile = blockIdx.x * 16;
    int ntile = wave * 16;
    int half  = lane >> 4;                // 0 or 1: which K pair this lane holds
    int mrow  = mtile + (lane & 15);
    if (mrow >= n) mrow = n - 1;          // clamp loads; EXEC stays all-1 for WMMA
    int ncol  = ntile + (lane & 15);

    v8f acc = {};
    for (int k = 0; k < EMB; k += 4) {
        int kk = k + half * 2;
        v2f a, b;
        a.x = x[(size_t)mrow * EMB + kk];
        a.y = x[(size_t)mrow * EMB + kk + 1];
        b.x = w[(size_t)ncol * EMB + kk];
        b.y = w[(size_t)ncol * EMB + kk + 1];
        acc = __builtin_amdgcn_wmma_f32_16x16x4_f32(
            /*neg_a=*/false, a, /*neg_b=*/false, b,
            /*c_mod=*/(short)0, acc, /*reuse_a=*/false, /*reuse_b=*/false);
    }
    float bb = bias[ncol];
    #pragma unroll
    for (int j = 0; j < 8; ++j) {
        int m = mtile + j + half * 8;
        if (m < n) out[(size_t)m * EMB + ncol] = acc[j] + bb;
    }
}

extern "C" void kernel_launch(void* const* d_in, const int* in_sizes, int n_in,
                              void* d_out, int out_size, void* d_ws, size_t ws_size,
                              hipStream_t stream) {
    const int*   xi  = (const int*)d_in[0];     // x_indices [N]
    const int*   ei  = (const int*)d_in[1];     // edge_index [2, E] flat
    const float* emb = (const float*)d_in[2];   // [N, 64]
    const float* lw  = (const float*)d_in[3];   // [64, 64]
    const float* lb  = (const float*)d_in[4];   // [64]
    float* out = (float*)d_out;                 // [N, 64]

    int n = in_sizes[0];
    int e = in_sizes[1] / 2;
    const int* row = ei;
    const int* col = ei + e;

    // Workspace: dinv [N floats] then ping buffer [N*64 floats] (~26 MB).
    float* dinv = (float*)d_ws;
    size_t off  = (((size_t)n * sizeof(float)) + 511) & ~(size_t)511;
    float* bufA = (float*)((char*)d_ws + off);
    // d_out doubles as the second ping-pong buffer (fully overwritten later).

    const int BT = 256;
    int gN   = (n + BT - 1) / BT;                      // 1 thread / node
    int gE   = (e + BT - 1) / BT;                      // 1 thread / edge
    long long tN32 = (long long)n * 32;                // 1 wave / node
    long long tE32 = (long long)e * 32;                // 1 wave / edge
    int gN32 = (int)((tN32 + BT - 1) / BT);
    int gE32 = (int)((tE32 + BT - 1) / BT);

    // Degree + dinv
    k_deg_init<<<gN, BT, 0, stream>>>(dinv, n);
    k_deg_count<<<gE, BT, 0, stream>>>(col, dinv, e);
    k_dinv<<<gN, BT, 0, stream>>>(dinv, n);

    // Hop 1: emb[xi] -> bufA
    k_hop_init <<<gN32, BT, 0, stream>>>(emb, xi, dinv, bufA, n);
    k_hop_edges<<<gE32, BT, 0, stream>>>(row, col, dinv, emb, xi, bufA, e);
    // Hop 2: bufA -> d_out (scratch)
    k_hop_init <<<gN32, BT, 0, stream>>>(bufA, nullptr, dinv, out, n);
    k_hop_edges<<<gE32, BT, 0, stream>>>(row, col, dinv, bufA, nullptr, out, e);
    // Hop 3: d_out -> bufA
    k_hop_init <<<gN32, BT, 0, stream>>>(out, nullptr, dinv, bufA, n);
    k_hop_edges<<<gE32, BT, 0, stream>>>(row, col, dinv, out, nullptr, bufA, e);

    // Final linear via WMMA: bufA @ lw^T + lb -> d_out
    int gG = (n + 15) / 16;
    k_linear_wmma<<<gG, 128, 0, stream>>>(bufA, lw, lb, out, n);
}